// SGRUCell_69372311765504
// MI455X (gfx1250) — compile-verified
//
#include <hip/hip_runtime.h>

#define TT 64
#define BB 32
#define INP 128
#define HH 256
#define RR 32
#define H2 512
#define CLIPV 1.0f
#define LN_EPS 1e-5f

typedef float f4  __attribute__((ext_vector_type(4)));
typedef float v2f __attribute__((ext_vector_type(2)));
typedef float v8f __attribute__((ext_vector_type(8)));

__device__ __forceinline__ float sigm(float x) { return 1.0f / (1.0f + expf(-x)); }
__device__ __forceinline__ float softpl(float x) { return (x > 20.0f) ? x : log1pf(expf(x)); }

// ---------------------------------------------------------------------------
// Init: copy dU0/tE0 into carry regions (in d_out), compute plast0 = sp*dU0.h0
// one wave32 per (b,i) row; b128 vectorized.
// ---------------------------------------------------------------------------
__global__ void k_init(const float* __restrict__ dU0, const float* __restrict__ tE0,
                       const float* __restrict__ h0, const float* __restrict__ alpha,
                       float* __restrict__ dU, float* __restrict__ tE,
                       float* __restrict__ plast) {
  const int lane = threadIdx.x & 31;
  const int row  = blockIdx.x * 8 + (threadIdx.x >> 5);   // 0..B*H-1
  const int b    = row >> 8;
  const float sp = softpl(alpha[0]);
  const int base = row * HH;
  const float* hrow = h0 + b * HH;
  float acc = 0.0f;
#pragma unroll
  for (int p = 0; p < 2; ++p) {
    const int j = p * 128 + lane * 4;
    f4 d = *(const f4*)(dU0 + base + j);
    f4 e = *(const f4*)(tE0 + base + j);
    f4 h = *(const f4*)(hrow + j);
    *(f4*)(dU + base + j) = d;
    *(f4*)(tE + base + j) = e;
    acc += d[0]*h[0] + d[1]*h[1] + d[2]*h[2] + d[3]*h[3];
  }
#pragma unroll
  for (int o = 16; o > 0; o >>= 1) acc += __shfl_xor(acc, o, 32);
  if (lane == 0) plast[row] = sp * acc;
}

// ---------------------------------------------------------------------------
// Precompute clip planes: upper/lower from W_dv = Wh2h[H:2H,:]
// ---------------------------------------------------------------------------
__global__ void k_ul(const float* __restrict__ Wh2h, const float* __restrict__ alpha,
                     float* __restrict__ upper, float* __restrict__ lower) {
  const int idx = blockIdx.x * blockDim.x + threadIdx.x;       // i*H + j
  const float denom = softpl(alpha[0]) + 1e-8f;
  const float w = Wh2h[HH * HH + idx];                          // Wh2h[(H+i)*H + j]
  upper[idx] =  fmaxf(CLIPV - w, 0.0f) / denom;
  lower[idx] = -fmaxf(CLIPV + w, 0.0f) / denom;
}

// ---------------------------------------------------------------------------
// fp32 WMMA GEMM: C[M,N] = A[M,K] @ W[N,K]^T + bias[N]
// one wave per 16x16 C tile; K-loop of V_WMMA_F32_16X16X4_F32.
// A frag (16x4):  lane<16 -> M=lane,    K = k..k+1 ; lane>=16 -> M=lane-16, K = k+2..k+3
// B frag (4x16):  lane<16 -> N=lane,    K = k..k+1 ; lane>=16 -> N=lane-16, K = k+2..k+3
// D (16x16):      vgpr r  -> M = r (+8 for upper lanes), N = lane&15
// ---------------------------------------------------------------------------
__global__ void k_gemm_wmma(const float* __restrict__ A, const float* __restrict__ W,
                            const float* __restrict__ bias, float* __restrict__ C,
                            int M, int N, int K) {
  const int lane  = threadIdx.x & 31;
  const int wave  = (blockIdx.x * blockDim.x + threadIdx.x) >> 5;
  const int ntile = N >> 4;
  const int mt = wave / ntile;
  const int nt = wave - mt * ntile;
  const int m0 = mt << 4, n0 = nt << 4;
  const int l15   = lane & 15;
  const int khalf = (lane >> 4) << 1;
  const float* arow = A + (size_t)(m0 + l15) * K;
  const float* wrow = W + (size_t)(n0 + l15) * K;
  v8f c = {0.f, 0.f, 0.f, 0.f, 0.f, 0.f, 0.f, 0.f};
  for (int k = 0; k < K; k += 4) {
    v2f a, b;
    a.x = arow[k + khalf]; a.y = arow[k + khalf + 1];
    b.x = wrow[k + khalf]; b.y = wrow[k + khalf + 1];
    c = __builtin_amdgcn_wmma_f32_16x16x4_f32(false, a, false, b, (short)0, c,
                                              false, false);
  }
  const int col   = n0 + l15;
  const int rbase = m0 + ((lane >> 4) << 3);
  const float bv = bias[col];
#pragma unroll
  for (int r = 0; r < 8; ++r)
    C[(size_t)(rbase + r) * N + col] = c[r] + bv;
}

// ---------------------------------------------------------------------------
// LayerNorm over rows of 512 (for precomputed Wx), in place.
// ---------------------------------------------------------------------------
__global__ void k_ln512(float* __restrict__ X, const float* __restrict__ g,
                        const float* __restrict__ b) {
  __shared__ float r1[256];
  __shared__ float r2[256];
  float* x = X + (size_t)blockIdx.x * H2;
  const int tid = threadIdx.x;
  float x0 = x[tid], x1 = x[tid + 256];
  r1[tid] = x0 + x1;
  r2[tid] = x0 * x0 + x1 * x1;
  __syncthreads();
  for (int s = 128; s > 0; s >>= 1) {
    if (tid < s) { r1[tid] += r1[tid + s]; r2[tid] += r2[tid + s]; }
    __syncthreads();
  }
  const float mean = r1[0] * (1.0f / 512.0f);
  const float var  = r2[0] * (1.0f / 512.0f) - mean * mean;
  const float rstd = rsqrtf(var + LN_EPS);
  x[tid]       = (x0 - mean) * rstd * g[tid]       + b[tid];
  x[tid + 256] = (x1 - mean) * rstd * g[tid + 256] + b[tid + 256];
}

// ---------------------------------------------------------------------------
// Per-step small fused kernel: one block per batch row.
// Wh += plast (upper half); LN; pre = Wx+Wh; z/dv; v,nh; mod MLP; te update.
// ---------------------------------------------------------------------------
__global__ void k_step(const float* __restrict__ Wx_t, const float* __restrict__ WhRaw,
                       const float* __restrict__ plast,
                       const float* __restrict__ lnh_g, const float* __restrict__ lnh_b,
                       const float* __restrict__ v_in, float* __restrict__ v_out,
                       const float* __restrict__ h_prev, const float* __restrict__ te_prev,
                       const float* __restrict__ Wh2mod, const float* __restrict__ bh2mod,
                       const float* __restrict__ Wmod2h, const float* __restrict__ bmod2h,
                       float* __restrict__ outs_t, float* __restrict__ mods_t,
                       float* __restrict__ keys_t, float* __restrict__ h_out,
                       float* __restrict__ te_out) {
  __shared__ float r1[512];
  __shared__ float r2[512];
  __shared__ float sh_pre[512];
  __shared__ float sh_nh[256];
  __shared__ float sh_mod1[32];
  __shared__ float sh_mod[4];
  const int b = blockIdx.x, tid = threadIdx.x;

  float wh = WhRaw[b * H2 + tid];
  if (tid >= HH) wh += plast[b * HH + (tid - HH)];   // plast already * softplus(alpha)
  r1[tid] = wh; r2[tid] = wh * wh;
  __syncthreads();
  for (int s = 256; s > 0; s >>= 1) {
    if (tid < s) { r1[tid] += r1[tid + s]; r2[tid] += r2[tid + s]; }
    __syncthreads();
  }
  const float mean = r1[0] * (1.0f / 512.0f);
  const float var  = r2[0] * (1.0f / 512.0f) - mean * mean;
  const float rstd = rsqrtf(var + LN_EPS);
  const float whln = (wh - mean) * rstd * lnh_g[tid] + lnh_b[tid];
  sh_pre[tid] = Wx_t[b * H2 + tid] + whln;
  __syncthreads();

  if (tid < HH) {
    float z  = sigm(sh_pre[tid]);
    float dv = sh_pre[tid + HH];
    float vn = (1.0f - z) * v_in[b * HH + tid] + z * dv;
    float nh = fmaxf(vn, 0.0f);
    v_out[b * HH + tid]  = vn;
    sh_nh[tid]           = nh;
    outs_t[b * HH + tid] = nh;
    h_out[b * HH + tid]  = nh;
  }
  __syncthreads();
  if (tid < RR) {
    float acc = bh2mod[tid];
    const float* wr = Wh2mod + tid * HH;
    for (int j = 0; j < HH; ++j) acc += wr[j] * sh_nh[j];
    sh_mod1[tid] = fmaxf(acc, 0.0f);
  }
  __syncthreads();
  if (tid < 3) {
    float acc = bmod2h[tid];
    const float* wr = Wmod2h + tid * RR;
    for (int r = 0; r < RR; ++r) acc += wr[r] * sh_mod1[r];
    sh_mod[tid] = acc;
    mods_t[b * 3 + tid] = acc;
  }
  __syncthreads();
  const float rg = sigm(sh_mod[0]);
  if (tid < HH) {
    float ten = (1.0f - rg) * te_prev[b * HH + tid] + rg * h_prev[b * HH + tid];
    keys_t[b * HH + tid] = ten;
    te_out[b * HH + tid] = ten;
  }
}

// ---------------------------------------------------------------------------
// Big per-step kernel: one wave32 per (b,i) row of B x H x H.
// skew / tE_n / dU_n + clip; dU,tE stay L2-resident; dicts streamed NT;
// fused plast for next step with wave reduction.
// ---------------------------------------------------------------------------
__global__ void k_big(const float* __restrict__ nh_t, const float* __restrict__ te_t,
                      const float* __restrict__ mods_t,
                      const float* __restrict__ alpha, const float* __restrict__ tauU,
                      float* __restrict__ dU, float* __restrict__ tE,
                      const float* __restrict__ upper, const float* __restrict__ lower,
                      float* __restrict__ dicts_t, float* __restrict__ plast) {
  __shared__ __align__(16) float s_nh[HH];
  __shared__ __align__(16) float s_te[HH];
  const int lane = threadIdx.x & 31;
  const int wave = threadIdx.x >> 5;
  const int row  = blockIdx.x * 8 + wave;       // rows within a block share b
  const int b    = row >> 8;
  const int i    = row & 255;
  if (threadIdx.x < HH) {
    s_nh[threadIdx.x] = nh_t[b * HH + threadIdx.x];
    s_te[threadIdx.x] = te_t[b * HH + threadIdx.x];
  }
  __syncthreads();
  const float sg  = sigm(mods_t[b * 3 + 1]);
  const float mg  = mods_t[b * 3 + 2];
  const float tau = sigm(tauU[0]);
  const float sp  = softpl(alpha[0]);
  const float nhi = s_nh[i], tei = s_te[i];
  const int base  = row * HH;
  const int ubase = i * HH;
  float acc = 0.0f;
#pragma unroll
  for (int p = 0; p < 2; ++p) {
    const int j = p * 128 + lane * 4;
    f4 e   = *(const f4*)(tE + base + j);
    f4 d   = *(const f4*)(dU + base + j);
    f4 up  = *(const f4*)(upper + ubase + j);
    f4 lo  = *(const f4*)(lower + ubase + j);
    f4 nh4 = *(const f4*)(s_nh + j);
    f4 te4 = *(const f4*)(s_te + j);
    f4 en, dn;
#pragma unroll
    for (int c = 0; c < 4; ++c) {
      float nhj  = nh4[c], tej = te4[c];
      float skew = nhi * tej - tei * nhj;
      float tEn  = (1.0f - sg) * e[c] + sg * skew;
      float dUn  = (1.0f - tau) * d[c] + tau * mg * tEn;
      dUn = fminf(dUn, up[c]);
      dUn = fmaxf(dUn, lo[c]);
      en[c] = tEn;
      dn[c] = dUn;
      acc += dUn * nhj;
    }
    *(f4*)(tE + base + j) = en;
    *(f4*)(dU + base + j) = dn;
    __builtin_nontemporal_store(dn, (f4*)(dicts_t + base + j));  // stream, keep L2 for dU/tE
  }
#pragma unroll
  for (int o = 16; o > 0; o >>= 1) acc += __shfl_xor(acc, o, 32);
  if (lane == 0) plast[row] = sp * acc;
}

// ---------------------------------------------------------------------------
extern "C" void kernel_launch(void* const* d_in, const int* in_sizes, int n_in,
                              void* d_out, int out_size, void* d_ws, size_t ws_size,
                              hipStream_t stream) {
  (void)in_sizes; (void)n_in; (void)out_size; (void)ws_size;
  const float* x      = (const float*)d_in[0];
  const float* h0     = (const float*)d_in[1];
  const float* v0     = (const float*)d_in[2];
  const float* dU0    = (const float*)d_in[3];
  const float* te0    = (const float*)d_in[4];
  const float* tE0    = (const float*)d_in[5];
  const float* Wx2h   = (const float*)d_in[6];
  const float* bx2h   = (const float*)d_in[7];
  const float* lnx_g  = (const float*)d_in[8];
  const float* lnx_b  = (const float*)d_in[9];
  const float* Wh2h   = (const float*)d_in[10];
  const float* bh2h   = (const float*)d_in[11];
  const float* lnh_g  = (const float*)d_in[12];
  const float* lnh_b  = (const float*)d_in[13];
  const float* Wh2mod = (const float*)d_in[14];
  const float* bh2mod = (const float*)d_in[15];
  const float* Wmod2h = (const float*)d_in[16];
  const float* bmod2h = (const float*)d_in[17];
  const float* alpha  = (const float*)d_in[18];
  const float* tauU   = (const float*)d_in[19];

  float* out = (float*)d_out;
  const size_t BH  = (size_t)BB * HH;        // 8192
  const size_t BHH = (size_t)BB * HH * HH;   // 2,097,152
  float* o_v     = out;                      // (B,H)
  float* o_h     = o_v + BH;                 // (B,H)
  float* o_dU    = o_h + BH;                 // (B,H,H) -- doubles as dU carry
  float* o_te    = o_dU + BHH;               // (B,H)
  float* o_tE    = o_te + BH;                // (B,H,H) -- doubles as tE carry
  float* o_outs  = o_tE + BHH;               // (T,B,H)
  float* o_mods  = o_outs + (size_t)TT * BH; // (T,B,3)
  float* o_keys  = o_mods + (size_t)TT * BB * 3; // (T,B,H)
  float* o_dicts = o_keys + (size_t)TT * BH; // (T,B,H,H)

  float* ws    = (float*)d_ws;
  float* WxAll = ws;                               // T*B*512  (4 MB)
  float* WhRaw = WxAll + (size_t)TT * BB * H2;     // B*512
  float* plast = WhRaw + (size_t)BB * H2;          // B*H
  float* upper = plast + BH;                       // H*H
  float* lower = upper + (size_t)HH * HH;          // H*H

  // one-time preamble
  k_init<<<dim3(BB * HH / 8), dim3(256), 0, stream>>>(dU0, tE0, h0, alpha,
                                                      o_dU, o_tE, plast);
  k_ul<<<dim3(HH * HH / 256), dim3(256), 0, stream>>>(Wh2h, alpha, upper, lower);
  k_gemm_wmma<<<dim3(((TT * BB / 16) * (H2 / 16)) / 8), dim3(256), 0, stream>>>(
      x, Wx2h, bx2h, WxAll, TT * BB, H2, INP);
  k_ln512<<<dim3(TT * BB), dim3(256), 0, stream>>>(WxAll, lnx_g, lnx_b);

  // sequential recurrence
  for (int t = 0; t < TT; ++t) {
    const float* hprev  = (t == 0) ? h0  : (o_outs + (size_t)(t - 1) * BH);
    const float* teprev = (t == 0) ? te0 : (o_keys + (size_t)(t - 1) * BH);
    const float* vin    = (t == 0) ? v0  : o_v;
    k_gemm_wmma<<<dim3(((BB / 16) * (H2 / 16)) / 8), dim3(256), 0, stream>>>(
        hprev, Wh2h, bh2h, WhRaw, BB, H2, HH);
    k_step<<<dim3(BB), dim3(512), 0, stream>>>(
        WxAll + (size_t)t * BB * H2, WhRaw, plast, lnh_g, lnh_b, vin, o_v,
        hprev, teprev, Wh2mod, bh2mod, Wmod2h, bmod2h,
        o_outs + (size_t)t * BH, o_mods + (size_t)t * BB * 3,
        o_keys + (size_t)t * BH, o_h, o_te);
    k_big<<<dim3(BB * HH / 8), dim3(256), 0, stream>>>(
        o_outs + (size_t)t * BH, o_keys + (size_t)t * BH,
        o_mods + (size_t)t * BB * 3, alpha, tauU, o_dU, o_tE, upper, lower,
        o_dicts + (size_t)t * BHH, plast);
  }
}